// SelfAttention_50053548868075
// MI455X (gfx1250) — compile-verified
//
#include <hip/hip_runtime.h>
#include <math.h>

typedef float v2f __attribute__((ext_vector_type(2)));
typedef float v8f __attribute__((ext_vector_type(8)));

#define BB 4
#define SS 4096
#define DD 128
#define KSPLIT 4                      // key-dimension split for occupancy
#define KCHUNK (SS / 16 / KSPLIT)     // 64 key blocks per chunk
#define QK_SCALE 0.08838834764831845f // 1/sqrt(128)

__device__ __forceinline__ v8f wmma_f32(v2f a, v2f b, v8f c) {
    // V_WMMA_F32_16X16X4_F32 : D = A(16x4) * B(4x16) + C(16x16)
    return __builtin_amdgcn_wmma_f32_16x16x4_f32(
        /*neg_a=*/false, a, /*neg_b=*/false, b,
        /*c_mod=*/(short)0, c, /*reuse_a=*/false, /*reuse_b=*/false);
}

// ---------------------------------------------------------------------------
// Kernel 1: q = (x*m) @ Wq * scale, k = (x*m) @ Wk, v = (x*m) @ Wv
// One wave per 16x16 output tile. job = (mtile, mat, ntile). 24576 waves.
// ---------------------------------------------------------------------------
__global__ __launch_bounds__(128) void qkv_kernel(
    const float* __restrict__ x, const int* __restrict__ mask,
    const float* __restrict__ Wq, const float* __restrict__ Wk,
    const float* __restrict__ Wv,
    float* __restrict__ qb, float* __restrict__ kb, float* __restrict__ vb) {
    int lane = threadIdx.x & 31;
    int wave = threadIdx.x >> 5;
    int job  = blockIdx.x * (blockDim.x >> 5) + wave;   // [0, 1024*24)
    int mt   = job / 24;
    int rem  = job % 24;
    int mat  = rem >> 3;         // 0=q, 1=k, 2=v
    int nt   = rem & 7;          // output column tile
    const float* W  = (mat == 0) ? Wq : (mat == 1) ? Wk : Wv;
    float*       Ob = (mat == 0) ? qb : (mat == 1) ? kb : vb;

    int hf = lane >> 4;          // wave half: 0/1
    int l16 = lane & 15;
    int khalf = hf << 1;         // K offset 0 or 2

    long arow = (long)mt * 16 + l16;
    float mr  = (float)mask[arow];           // multiplicative input mask
    const float* Ap = x + arow * DD;
    const float* Bp = W + nt * 16 + l16;     // column (nt*16 + l16)

    v8f acc0 = {0.f,0.f,0.f,0.f,0.f,0.f,0.f,0.f};
    v8f acc1 = {0.f,0.f,0.f,0.f,0.f,0.f,0.f,0.f};
#pragma unroll
    for (int k0 = 0; k0 < DD; k0 += 8) {    // two independent WMMA chains
        v2f a0 = *(const v2f*)(Ap + k0 + khalf);
        v2f a1 = *(const v2f*)(Ap + k0 + 4 + khalf);
        a0 *= mr;
        a1 *= mr;
        v2f b0, b1;
        b0.x = Bp[(long)(k0 + khalf) * DD];
        b0.y = Bp[(long)(k0 + khalf + 1) * DD];
        b1.x = Bp[(long)(k0 + 4 + khalf) * DD];
        b1.y = Bp[(long)(k0 + 4 + khalf + 1) * DD];
        acc0 = wmma_f32(a0, b0, acc0);
        acc1 = wmma_f32(a1, b1, acc1);
    }
    v8f acc = acc0 + acc1;
    float sc = (mat == 0) ? QK_SCALE : 1.0f; // fold 1/sqrt(D) into q
    int col = nt * 16 + l16;
    long rbase = (long)mt * 16 + hf * 8;
#pragma unroll
    for (int r = 0; r < 8; ++r)
        Ob[(rbase + r) * DD + col] = acc[r] * sc;
}

// ---------------------------------------------------------------------------
// score tile helper: S(16x16) = Q_tile . K_tile^T with dual WMMA chains
// ---------------------------------------------------------------------------
__device__ __forceinline__ v8f score_tile(const float* __restrict__ Ap,
                                          const float* __restrict__ Bpr,
                                          int khalf) {
    v8f s0 = {0.f,0.f,0.f,0.f,0.f,0.f,0.f,0.f};
    v8f s1 = {0.f,0.f,0.f,0.f,0.f,0.f,0.f,0.f};
#pragma unroll
    for (int d0 = 0; d0 < DD; d0 += 8) {
        v2f a0 = *(const v2f*)(Ap + d0 + khalf);
        v2f b0 = *(const v2f*)(Bpr + d0 + khalf);
        v2f a1 = *(const v2f*)(Ap + d0 + 4 + khalf);
        v2f b1 = *(const v2f*)(Bpr + d0 + 4 + khalf);
        s0 = wmma_f32(a0, b0, s0);
        s1 = wmma_f32(a1, b1, s1);
    }
    return s0 + s1;
}

// ---------------------------------------------------------------------------
// Kernel 2: partial softmax stats (max,sum) per 1024-key chunk.
// wave id t: chunk = t&3, qtile = t>>2.  4096 waves.
// ---------------------------------------------------------------------------
__global__ __launch_bounds__(128) void attn_pass1(
    const float* __restrict__ qb, const float* __restrict__ kb,
    const int* __restrict__ mask,
    float* __restrict__ Mpart, float* __restrict__ Lpart) {
    int lane = threadIdx.x & 31;
    int wave = threadIdx.x >> 5;
    int t    = blockIdx.x * (blockDim.x >> 5) + wave;   // [0, 4096)
    int chunk = t & (KSPLIT - 1);
    int tile  = t >> 2;              // [0, 1024)
    int bidx  = tile >> 8;
    int qt    = tile & 255;
    int hf = lane >> 4, l16 = lane & 15, khalf = hf << 1;

    long qrow0 = (long)bidx * SS + (long)qt * 16;
    const float* Ap = qb + (qrow0 + l16) * DD;
    const float* Kb = kb + (long)bidx * SS * DD;

    float mv[8], mrun[8], lrun[8];
#pragma unroll
    for (int r = 0; r < 8; ++r) {
        mv[r]   = (float)mask[qrow0 + hf * 8 + r];  // row mask (scores, pre-softmax)
        mrun[r] = -1e30f;
        lrun[r] = 0.0f;
    }

    int kt0 = chunk * KCHUNK;
    for (int kt = kt0; kt < kt0 + KCHUNK; ++kt) {
        v8f s = score_tile(Ap, Kb + (long)(kt * 16 + l16) * DD, khalf);
#pragma unroll
        for (int r = 0; r < 8; ++r) {
            float v  = s[r] * mv[r];
            float vm = v;
            vm = fmaxf(vm, __shfl_xor(vm, 1));
            vm = fmaxf(vm, __shfl_xor(vm, 2));
            vm = fmaxf(vm, __shfl_xor(vm, 4));
            vm = fmaxf(vm, __shfl_xor(vm, 8)); // row max within 16-lane group
            float mnew = fmaxf(mrun[r], vm);
            float e = __expf(v - mnew);
            e += __shfl_xor(e, 1);
            e += __shfl_xor(e, 2);
            e += __shfl_xor(e, 4);
            e += __shfl_xor(e, 8);             // row sum within 16-lane group
            lrun[r] = lrun[r] * __expf(mrun[r] - mnew) + e;
            mrun[r] = mnew;
        }
    }
    if (l16 == 0) {
        long base = (long)chunk * (BB * SS) + qrow0;
#pragma unroll
        for (int r = 0; r < 8; ++r) {
            Mpart[base + hf * 8 + r] = mrun[r];
            Lpart[base + hf * 8 + r] = lrun[r];
        }
    }
}

// ---------------------------------------------------------------------------
// Kernel 3: combine per-chunk (m,l) -> exact (m,l) per row
// ---------------------------------------------------------------------------
__global__ void combine_ml(const float* __restrict__ Mpart,
                           const float* __restrict__ Lpart,
                           float* __restrict__ Mrow, float* __restrict__ Lrow) {
    int i = blockIdx.x * blockDim.x + threadIdx.x;
    if (i >= BB * SS) return;
    float m = -1e30f;
#pragma unroll
    for (int c = 0; c < KSPLIT; ++c)
        m = fmaxf(m, Mpart[(long)c * (BB * SS) + i]);
    float l = 0.0f;
#pragma unroll
    for (int c = 0; c < KSPLIT; ++c)
        l += Lpart[(long)c * (BB * SS) + i] * __expf(Mpart[(long)c * (BB * SS) + i] - m);
    Mrow[i] = m;
    Lrow[i] = l;
}

// ---------------------------------------------------------------------------
// Kernel 4: exact-normalized P over a key chunk, colsum atomics,
// partial O = P@V accumulated into obuf via atomicAdd. 4096 waves.
// ---------------------------------------------------------------------------
__global__ __launch_bounds__(128) void attn_pass2(
    const float* __restrict__ qb, const float* __restrict__ kb,
    const float* __restrict__ vb, const int* __restrict__ mask,
    const float* __restrict__ Mrow, const float* __restrict__ Lrow,
    float* __restrict__ colsum, float* __restrict__ obuf) {
    __shared__ __align__(16) float lds[4 * 16 * 20];   // 4 waves * 16x16 P, row stride 20
    int lane = threadIdx.x & 31;
    int wave = threadIdx.x >> 5;
    int t    = blockIdx.x * (blockDim.x >> 5) + wave;
    int chunk = t & (KSPLIT - 1);
    int tile  = t >> 2;
    int bidx  = tile >> 8, qt = tile & 255;
    int hf = lane >> 4, l16 = lane & 15, khalf = hf << 1;

    long qrow0 = (long)bidx * SS + (long)qt * 16;
    const float* Ap = qb + (qrow0 + l16) * DD;
    const float* Kb = kb + (long)bidx * SS * DD;
    const float* Vb = vb + (long)bidx * SS * DD;
    float* P = lds + wave * (16 * 20);

    float mv[8], mf[8], li[8];
#pragma unroll
    for (int r = 0; r < 8; ++r) {
        long rr = qrow0 + hf * 8 + r;
        mv[r] = (float)mask[rr];
        mf[r] = Mrow[rr];
        li[r] = 1.0f / Lrow[rr];
    }
    v8f acc[8];
#pragma unroll
    for (int n = 0; n < 8; ++n) acc[n] = (v8f){0.f,0.f,0.f,0.f,0.f,0.f,0.f,0.f};

    int kt0 = chunk * KCHUNK;
    for (int kt = kt0; kt < kt0 + KCHUNK; ++kt) {
        // ---- scores: S = q . k^T (q pre-scaled by 1/sqrt(D))
        v8f s = score_tile(Ap, Kb + (long)(kt * 16 + l16) * DD, khalf);
        __syncthreads();   // previous iteration's P reads complete (uniform trips)
        float cs = 0.0f;
#pragma unroll
        for (int r = 0; r < 8; ++r) {
            float p = __expf(s[r] * mv[r] - mf[r]) * li[r];
            cs += p;
            P[(hf * 8 + r) * 20 + l16] = p;    // P[q][k]
        }
        cs += __shfl_xor(cs, 16);              // combine the two halves
        if (hf == 0)
            atomicAdd(&colsum[(long)bidx * SS + kt * 16 + l16], cs);
        __syncthreads();   // P visible for A-fragment reads
        // ---- O += P @ V  (8 independent WMMA chains, one per n-tile)
        const float* Vr = Vb + (long)(kt * 16 + khalf) * DD + l16;
#pragma unroll
        for (int n = 0; n < 8; ++n) {
#pragma unroll
            for (int kk = 0; kk < 16; kk += 4) {
                v2f a = *(const v2f*)(P + l16 * 20 + kk + khalf);
                v2f bf;
                bf.x = Vr[(long)kk * DD + n * 16];
                bf.y = Vr[(long)(kk + 1) * DD + n * 16];
                acc[n] = wmma_f32(a, bf, acc[n]);
            }
        }
    }
    // accumulate partial output tile (key-split) into obuf
#pragma unroll
    for (int n = 0; n < 8; ++n) {
#pragma unroll
        for (int r = 0; r < 8; ++r) {
            long row = qrow0 + hf * 8 + r;
            atomicAdd(&obuf[row * DD + n * 16 + l16], acc[n][r]);
        }
    }
}

// ---------------------------------------------------------------------------
// helpers: zero accumulators each call (graph-replay determinism),
// finalize out = obuf * rowmask, aw = colsum/S * mask
// ---------------------------------------------------------------------------
__global__ void zero_accum(float* __restrict__ p, long n) {
    long i = (long)blockIdx.x * blockDim.x + threadIdx.x;
    if (i < n) p[i] = 0.0f;
}

__global__ void finalize_out(const float* __restrict__ obuf,
                             const int* __restrict__ mask,
                             float* __restrict__ out) {
    long i = (long)blockIdx.x * blockDim.x + threadIdx.x;
    if (i < (long)BB * SS * DD) {
        long row = i >> 7;                   // / DD
        out[i] = obuf[i] * (float)mask[row];
    }
}

__global__ void finalize_aw(const float* __restrict__ colsum,
                            const int* __restrict__ mask,
                            float* __restrict__ aw) {
    int i = blockIdx.x * blockDim.x + threadIdx.x;
    if (i < BB * SS)
        aw[i] = colsum[i] * (1.0f / (float)SS) * (float)mask[i];
}

// ---------------------------------------------------------------------------
extern "C" void kernel_launch(void* const* d_in, const int* in_sizes, int n_in,
                              void* d_out, int out_size, void* d_ws, size_t ws_size,
                              hipStream_t stream) {
    const float* x   = (const float*)d_in[0];
    const int*   msk = (const int*)d_in[1];
    const float* Wq  = (const float*)d_in[2];
    const float* Wk  = (const float*)d_in[3];
    const float* Wv  = (const float*)d_in[4];

    float* ws = (float*)d_ws;
    const long NQ = (long)BB * SS * DD;      // 2,097,152 floats
    const long NR = (long)BB * SS;           // 16,384 rows
    float* qb    = ws;
    float* kb    = qb + NQ;
    float* vb    = kb + NQ;
    float* obuf  = vb + NQ;                  // fp32 output accumulator [B*S*D]
    float* csum  = obuf + NQ;                // column sums [B*S]  (contiguous w/ obuf)
    float* Mpart = csum + NR;                // [KSPLIT][B*S]
    float* Lpart = Mpart + (long)KSPLIT * NR;
    float* Mr    = Lpart + (long)KSPLIT * NR;
    float* Lr    = Mr + NR;

    float* out = (float*)d_out;              // [B,S,D]
    float* aw  = out + NQ;                   // [B,S]

    dim3 blk(128);
    // 1024 mtiles * 3 mats * 8 ntiles = 24576 waves / 4 per block
    qkv_kernel<<<dim3(24576 / 4), blk, 0, stream>>>(x, msk, Wq, Wk, Wv, qb, kb, vb);
    // zero obuf + csum in one sweep (they are contiguous)
    zero_accum<<<dim3((unsigned)((NQ + NR + 255) / 256)), dim3(256), 0, stream>>>(obuf, NQ + NR);
    attn_pass1<<<dim3(4096 / 4), blk, 0, stream>>>(qb, kb, msk, Mpart, Lpart);
    combine_ml<<<dim3((BB * SS + 255) / 256), dim3(256), 0, stream>>>(Mpart, Lpart, Mr, Lr);
    attn_pass2<<<dim3(4096 / 4), blk, 0, stream>>>(qb, kb, vb, msk, Mr, Lr, csum, obuf);
    finalize_out<<<dim3((unsigned)((NQ + 255) / 256)), dim3(256), 0, stream>>>(obuf, msk, out);
    finalize_aw<<<dim3((BB * SS + 255) / 256), dim3(256), 0, stream>>>(csum, msk, aw);
}